// BondCenteredTensorMomentDescriptor_62173946576949
// MI455X (gfx1250) — compile-verified
//
#include <hip/hip_runtime.h>
#include <math.h>

// ---------------------------------------------------------------------------
// BondCenteredTensorMomentDescriptor — MI455X (gfx1250).
//
// out_f (E x 50) = Z_f (E x 225) @ W_f (225 x 50), per f in [0,16)
//   Z_f[e, a*25+b] = bond[e,a,f] * y[e,b,f]
//   W_f[a*25+b, plane*25+cc] = realCG_{l1,l2,l3}[a',b',c'] * tp_w[p,f]
// via V_WMMA_F32_16X16X4_F32. 64 edges/block: each B fragment feeds 4 WMMAs
// (4 M-subtiles) to cut L2 fragment traffic 4x. Branch-free inner loop via
// zero-padded bond row (a_full==9).
// ---------------------------------------------------------------------------

typedef __attribute__((ext_vector_type(2))) float v2f;
typedef __attribute__((ext_vector_type(8))) float v8f;

#define KDIM 228            // 225 (9*25) padded to multiple of 4
#define NDIM 64             // 50 outputs padded to 4 WMMA n-tiles
#define NKT  57             // KDIM / 4
#define NF   16
#define EPB  64             // edges per block (4 M-subtiles of 16)

// ------------------------- CG math (device, fp64) --------------------------

__device__ __forceinline__ double dfac(int n) {
    const double t[13] = {1.,1.,2.,6.,24.,120.,720.,5040.,40320.,362880.,
                          3628800.,39916800.,479001600.};
    return t[n];
}

__device__ double cg_coef(int j1,int m1,int j2,int m2,int j3,int m3) {
    if (m1 + m2 != m3) return 0.0;
    int dlo = j1 - j2; if (dlo < 0) dlo = -dlo;
    if (j3 < dlo || j3 > j1 + j2) return 0.0;
    double pre = sqrt((2.0*j3+1.0)*dfac(j3+j1-j2)*dfac(j3-j1+j2)*dfac(j1+j2-j3)
                      / dfac(j1+j2+j3+1));
    pre *= sqrt(dfac(j3+m3)*dfac(j3-m3)*dfac(j1-m1)*dfac(j1+m1)
               *dfac(j2-m2)*dfac(j2+m2));
    int k0 = 0;
    if (j2-j3-m1 > k0) k0 = j2-j3-m1;
    if (j1-j3+m2 > k0) k0 = j1-j3+m2;
    int k1 = j1+j2-j3;
    if (j1-m1 < k1) k1 = j1-m1;
    if (j2+m2 < k1) k1 = j2+m2;
    double s = 0.0;
    for (int k = k0; k <= k1; ++k) {
        double d = dfac(k)*dfac(j1+j2-j3-k)*dfac(j1-m1-k)*dfac(j2+m2-k)
                 * dfac(j3-j2+m1+k)*dfac(j3-j1-m2+k);
        s += ((k & 1) ? -1.0 : 1.0) / d;
    }
    return pre * s;
}

__device__ void umat_entry(int l, int mr, int mc, double* re, double* im) {
    const double s2 = 0.7071067811865476;
    *re = 0.0; *im = 0.0;
    (void)l;
    if (mr > 0) {
        if (mc ==  mr) *re = ((mr & 1) ? -1.0 : 1.0) * s2;
        if (mc == -mr) *re = s2;
    } else if (mr == 0) {
        if (mc == 0) *re = 1.0;
    } else {
        int am = -mr;
        if (mc == mr) *im = s2;
        if (mc == am) *im = -(((am & 1) ? -1.0 : 1.0)) * s2;
    }
}

__device__ float real_cg_elem(int l1,int l2,int l3,int a,int b,int c) {
    int ma = a - l1, mb = b - l2, mc = c - l3;
    double gre = 0.0, gim = 0.0;
    int Ac[2]; int nA = (ma == 0) ? 1 : 2; Ac[0] = ma; Ac[1] = -ma;
    int Bc[2]; int nB = (mb == 0) ? 1 : 2; Bc[0] = mb; Bc[1] = -mb;
    int Cc[2]; int nC = (mc == 0) ? 1 : 2; Cc[0] = mc; Cc[1] = -mc;
    for (int ia = 0; ia < nA; ++ia)
    for (int ib = 0; ib < nB; ++ib) {
        int m1 = Ac[ia], m2 = Bc[ib], m3 = m1 + m2;
        if (m3 < -l3 || m3 > l3) continue;
        for (int ic = 0; ic < nC; ++ic) {
            if (Cc[ic] != m3) continue;
            double u1r,u1i,u2r,u2i,u3r,u3i;
            umat_entry(l1, ma, m1, &u1r, &u1i);
            umat_entry(l2, mb, m2, &u2r, &u2i);
            umat_entry(l3, mc, m3, &u3r, &u3i);
            u3i = -u3i;                       // conj(U3)
            double coef = cg_coef(l1,m1,l2,m2,l3,m3);
            double tr = u1r*u2r - u1i*u2i;
            double ti = u1r*u2i + u1i*u2r;
            double vr = tr*u3r - ti*u3i;
            double vi = tr*u3i + ti*u3r;
            gre += vr * coef;
            gim += vi * coef;
        }
    }
    return (float)(((l1 + l2 + l3) & 1) == 0 ? gre : gim);
}

__device__ int path_index(int l1,int l2,int l3) {
    int p = 0;
    for (int a1 = 0; a1 <= 2; ++a1)
        for (int a2 = 0; a2 <= 4; ++a2) {
            int lo = a1 - a2; if (lo < 0) lo = -lo;
            int hi = a1 + a2; if (hi > 4) hi = 4;
            for (int a3 = lo; a3 <= hi; ++a3) {
                if (a1 == l1 && a2 == l2 && a3 == l3) return p;
                ++p;
            }
        }
    return -1;
}

__device__ __forceinline__ int lblock25(int x) {
    return (x == 0) ? 0 : (x < 4 ? 1 : (x < 9 ? 2 : (x < 16 ? 3 : 4)));
}

// --------------------- Setup kernel A: dense CG table ----------------------

__global__ void setup_cg_kernel(float* __restrict__ cgval, int* __restrict__ pidx) {
    int t = blockIdx.x * blockDim.x + threadIdx.x;
    if (t >= KDIM * NDIM) return;
    int k = t >> 6, c = t & 63;
    float val = 0.0f; int p = -1;
    if (k < 225 && c < 50) {
        int af = k / 25, bf = k - af * 25;
        int l1 = (af == 0) ? 0 : (af < 4 ? 1 : 2);
        int l2 = lblock25(bf);
        int plane = c / 25, cc = c - plane * 25;
        int l3 = lblock25(cc);
        int lo = l1 - l2; if (lo < 0) lo = -lo;
        int hi = l1 + l2; if (hi > 4) hi = 4;
        if (l3 >= lo && l3 <= hi && (((l1 + l2 + l3) & 1) == plane)) {
            p = path_index(l1, l2, l3);
            val = real_cg_elem(l1, l2, l3, af - l1*l1, bf - l2*l2, cc - l3*l3);
        }
    }
    cgval[t] = val;
    pidx[t]  = p;
}

// ------ Setup kernel B: per-f WMMA B-fragments (pre-swizzled lane layout) --
// Assumed B layout (mirrors documented A 16x4): lane L holds column n=L%16,
// VGPR j holds K-row = j + 2*(L/16).
// frag float index = (((f*57 + kt)*4 + nt)*32 + lane)*2 + j

__global__ void setup_frag_kernel(float* __restrict__ frag,
                                  const float* __restrict__ cgval,
                                  const int*   __restrict__ pidx,
                                  const float* __restrict__ w) {
    int t = blockIdx.x * blockDim.x + threadIdx.x;
    const int total = NF * NKT * 4 * 32 * 2;
    if (t >= total) return;
    int j    = t & 1;
    int lane = (t >> 1) & 31;
    int nt   = (t >> 6) & 3;
    int fk   = t >> 8;            // f*57 + kt
    int kt   = fk % NKT;
    int f    = fk / NKT;
    int k = kt * 4 + j + 2 * (lane >> 4);
    int c = nt * 16 + (lane & 15);
    int i = k * 64 + c;
    int p = pidx[i];
    frag[t] = (p >= 0) ? cgval[i] * w[p * NF + f] : 0.0f;
}

// ------------------------------ Main kernel --------------------------------
// 256 threads (8 wave32), 64 edges per block.
// LDS strides odd mod 64 -> conflict-free across the 16 M-lanes.
// BSTRIDE covers a zeroed a_full==9 row so the K-tail needs no guard.

#define YSTRIDE 401   // 25*16 + 1
#define BSTRIDE 161   // 10*16 + 1 (row a=9 zero-filled)

__global__ __launch_bounds__(256)
void edge_tp_kernel(const float* __restrict__ desc,
                    const float* __restrict__ disp,
                    const int*   __restrict__ nidx,
                    const float* __restrict__ frag,
                    float*       __restrict__ out,
                    int E) {
    extern __shared__ float smem[];
    float* s_y = smem;                      // EPB * YSTRIDE
    float* s_b = smem + EPB * YSTRIDE;      // EPB * BSTRIDE

    const int tid = threadIdx.x;
    const int e0  = blockIdx.x * EPB;
    const int el4 = tid >> 4;      // 0..15
    const int f   = tid & 15;      // channel

    // ---------------- Phase 1: gather + bond into LDS ----------------
    #pragma unroll
    for (int ms = 0; ms < 4; ++ms) {
        const int el = ms * 16 + el4;
        const int edge = e0 + el;
        float* yb = s_y + el * YSTRIDE + f;
        float* bb = s_b + el * BSTRIDE + f;
        if (edge < E) {
            const int i0 = nidx[edge * 2 + 0];
            const int i1 = nidx[edge * 2 + 1];
            const float* p0 = desc + (size_t)i0 * 400 + f;
            const float* p1 = desc + (size_t)i1 * 400 + f;
            #pragma unroll
            for (int b = 0; b < 25; ++b) yb[b * 16] = p0[b * 16] + p1[b * 16];

            const float dx = disp[edge * 3 + 0];
            const float dy = disp[edge * 3 + 1];
            const float dz = disp[edge * 3 + 2];
            const float r  = sqrtf(dx*dx + dy*dy + dz*dz);
            const float inv = 1.0f / fmaxf(r, 1e-9f);
            const float ux = dx * inv, uy = dy * inv, uz = dz * inv;
            const float c1 = 0.4886025119029199f, c2 = 1.0925484305920792f;
            float sh[9];
            sh[0] = 0.28209479177387814f;
            sh[1] = c1 * uy;  sh[2] = c1 * uz;  sh[3] = c1 * ux;
            sh[4] = c2 * ux * uy;  sh[5] = c2 * uy * uz;
            sh[6] = 0.31539156525252005f * (3.0f * uz * uz - 1.0f);
            sh[7] = c2 * ux * uz;  sh[8] = 0.5f * c2 * (ux*ux - uy*uy);

            float rad = 0.0f;
            if (r < 5.0f) {
                const float px = 3.14159265358979323846f * (float)(f + 1) * (r * 0.2f);
                rad = (px > 1e-12f) ? (sinf(px) / px) : 1.0f;   // jnp.sinc
            }
            #pragma unroll
            for (int a = 0; a < 9; ++a) bb[a * 16] = rad * sh[a];
            bb[9 * 16] = 0.0f;                       // zero pad row (K tail)
        } else {
            #pragma unroll
            for (int b = 0; b < 25; ++b) yb[b * 16] = 0.0f;
            #pragma unroll
            for (int a = 0; a < 10; ++a) bb[a * 16] = 0.0f;
        }
    }
    __syncthreads();

    // ---------------- Phase 2: WMMA GEMM, 4 M-subtiles per B frag -----
    const int wave = tid >> 5;
    const int lane = tid & 31;
    const int m    = lane & 15;    // A row / D column
    const int half = lane >> 4;    // K pair select / D row-half

    for (int fi = 0; fi < 2; ++fi) {
        const int wf = wave + 8 * fi;
        const float* ybm[4];
        const float* bbm[4];
        #pragma unroll
        for (int mt = 0; mt < 4; ++mt) {
            ybm[mt] = s_y + (mt * 16 + m) * YSTRIDE + wf;
            bbm[mt] = s_b + (mt * 16 + m) * BSTRIDE + wf;
        }
        const float2* fb = (const float2*)frag;

        for (int nt = 0; nt < 4; ++nt) {
            v8f acc0 = {}, acc1 = {}, acc2 = {}, acc3 = {};
            const float2* fbase = fb + (size_t)((wf * NKT) * 4 + nt) * 32 + lane;
            for (int kt = 0; kt < NKT; ++kt) {
                const float2 bw = fbase[(size_t)kt * 128];
                v2f B; B.x = bw.x; B.y = bw.y;
                // shared per-lane K offsets (branch-free: a_full<=9 valid)
                const int k0 = kt * 4 + 2 * half;
                const int a0 = k0 / 25,      b0 = k0 - a0 * 25;
                const int k1 = k0 + 1;
                const int a1 = k1 / 25,      b1 = k1 - a1 * 25;
                const int ob0 = a0 * 16, oy0 = b0 * 16;
                const int ob1 = a1 * 16, oy1 = b1 * 16;
                v2f A;
                A.x = bbm[0][ob0] * ybm[0][oy0];
                A.y = bbm[0][ob1] * ybm[0][oy1];
                acc0 = __builtin_amdgcn_wmma_f32_16x16x4_f32(
                        false, A, false, B, (short)0, acc0, false, false);
                A.x = bbm[1][ob0] * ybm[1][oy0];
                A.y = bbm[1][ob1] * ybm[1][oy1];
                acc1 = __builtin_amdgcn_wmma_f32_16x16x4_f32(
                        false, A, false, B, (short)0, acc1, false, false);
                A.x = bbm[2][ob0] * ybm[2][oy0];
                A.y = bbm[2][ob1] * ybm[2][oy1];
                acc2 = __builtin_amdgcn_wmma_f32_16x16x4_f32(
                        false, A, false, B, (short)0, acc2, false, false);
                A.x = bbm[3][ob0] * ybm[3][oy0];
                A.y = bbm[3][ob1] * ybm[3][oy1];
                acc3 = __builtin_amdgcn_wmma_f32_16x16x4_f32(
                        false, A, false, B, (short)0, acc3, false, false);
            }
            // D layout: lane = column n, rows r + 8*half
            const int c_out = nt * 16 + m;
            if (c_out < 50) {
                #pragma unroll
                for (int r = 0; r < 8; ++r) {
                    const int er = r + 8 * half;
                    int eg;
                    eg = e0 +  0 + er; if (eg < E) out[(size_t)eg * 800 + c_out * 16 + wf] = acc0[r];
                    eg = e0 + 16 + er; if (eg < E) out[(size_t)eg * 800 + c_out * 16 + wf] = acc1[r];
                    eg = e0 + 32 + er; if (eg < E) out[(size_t)eg * 800 + c_out * 16 + wf] = acc2[r];
                    eg = e0 + 48 + er; if (eg < E) out[(size_t)eg * 800 + c_out * 16 + wf] = acc3[r];
                }
            }
        }
    }
}

// ------------------------------- Launcher ----------------------------------

extern "C" void kernel_launch(void* const* d_in, const int* in_sizes, int n_in,
                              void* d_out, int out_size, void* d_ws, size_t ws_size,
                              hipStream_t stream) {
    const float* desc = (const float*)d_in[0];   // (N_ATOMS,1,25,16) f32
    const float* tpw  = (const float*)d_in[1];   // (33,16) f32
    const float* disp = (const float*)d_in[2];   // (E,3) f32
    const int*   nidx = (const int*)  d_in[3];   // (E,2) i32
    float* out = (float*)d_out;                  // (E,2,25,16) f32
    const int E = in_sizes[2] / 3;

    // d_ws: [cgval f32 228*64][pidx i32 228*64][frag f32 16*57*4*64] ~1.05 MB
    char* ws = (char*)d_ws;
    float* cgval = (float*)ws;
    int*   pidx  = (int*)(ws + (size_t)KDIM * NDIM * 4);
    float* frag  = (float*)(ws + (size_t)KDIM * NDIM * 8);

    setup_cg_kernel<<<(KDIM * NDIM + 255) / 256, 256, 0, stream>>>(cgval, pidx);
    const int fragN = NF * NKT * 4 * 64;
    setup_frag_kernel<<<(fragN + 255) / 256, 256, 0, stream>>>(frag, cgval, pidx, tpw);

    const int ldsBytes = EPB * (YSTRIDE + BSTRIDE) * 4;   // 143,872 B
    static bool attrDone = false;   // idempotent, host-side only (not a work guard)
    if (!attrDone) {
        hipFuncSetAttribute((const void*)edge_tp_kernel,
                            hipFuncAttributeMaxDynamicSharedMemorySize, ldsBytes);
        attrDone = true;
    }
    edge_tp_kernel<<<(E + EPB - 1) / EPB, 256, ldsBytes, stream>>>(
        desc, disp, nidx, frag, out, E);
    (void)n_in; (void)out_size; (void)ws_size;
}